// IU_GCN_74646531605062
// MI455X (gfx1250) — compile-verified
//
#include <hip/hip_runtime.h>

// k-hop weighted SpMM for MI455X (gfx1250, wave32).
//   h <- scatter_sum(h[src] * w, dst), repeated 3 times.
//
// Memory-bound: working set (~115 MB: 3x 25.6MB feature buffers + 38.4MB edge
// stream, reused across 3 hops) fits in the 192MB L2. Limiter is L2 gather +
// atomic throughput, not HBM (23.3 TB/s) or FLOPs. So: default RT cache hints
// (everything wants L2 residency), fire-and-forget global_atomic_add_f32, and
// the per-edge scaling done exactly in f32 on the WMMA pipe with one-hot
// diagonal A matrices:
//   D(16x16) = A1(16x4) x B1(4x16) + A2(16x4) x B2(4x16)
//   A1[m][k] = (m==k)   ? w[k]   : 0   -> rows 0..3  = msg of edges 0..3
//   A2[m][k] = (m==8+k) ? w[4+k] : 0   -> rows 8..11 = msg of edges 4..7
// Rows r and r+8 share VGPR r (lanes 0-15 / 16-31), so the scatter runs with
// all 32 lanes active: no EXEC masking anywhere (WMMA requires EXEC all-1s).

#define N_NODES 100000
#define N_EDGES 3200000
#define D_FEAT  64

typedef __attribute__((ext_vector_type(2))) float v2f;
typedef __attribute__((ext_vector_type(8))) float v8f;

// Zero-fill with 128-bit stores (destinations are poisoned by the harness).
__global__ void gcn_zero_kernel(float4* __restrict__ p, int n4) {
    int i = blockIdx.x * blockDim.x + threadIdx.x;
    if (i < n4) p[i] = float4{0.f, 0.f, 0.f, 0.f};
}

// One hop. Each wave32 owns 8 consecutive edges; 4 feature tiles of 16.
// 256 threads/block = 8 waves = 64 edges/block; grid = E/64 blocks exactly,
// so the kernel is fully convergent.
__global__ __launch_bounds__(256) void gcn_hop_kernel(
    const float* __restrict__ hin,   // [N, 64]
    float*       __restrict__ hout,  // [N, 64], pre-zeroed
    const int*   __restrict__ src,   // [E]
    const int*   __restrict__ dst,   // [E]
    const float* __restrict__ ew)    // [E]
{
    const int  tid  = blockIdx.x * blockDim.x + threadIdx.x;
    const long base = (long)(tid >> 5) << 3;   // first of this wave's 8 edges
    if (base >= N_EDGES) return;               // wave-uniform; never taken

    const int lane = threadIdx.x & 31;
    const int m    = lane & 15;                // A/D row index, B/D col index
    const int hi   = lane >> 4;                // half-wave id
    const int k_x  = hi << 1;                  // K slot in .x lanes: 0 or 2
    const int k_y  = k_x + 1;                  // K slot in .y lanes: 1 or 3

    // --- A matrices: one-hot diagonals scaled by the edge weights (exact f32).
    // A layout (32-bit 16x4): lanes 0-15 hold K=0(.x),K=1(.y); lanes 16-31
    // hold K=2(.x),K=3(.y); row M = lane&15.
    const float w_a = ew[base + (m & 3)];      // edges 0..3 -> rows 0..3
    const float w_b = ew[base + 4 + (m & 3)];  // edges 4..7 -> rows 8..11
    v2f a1, a2;
    a1.x = (m == k_x)       ? w_a : 0.0f;
    a1.y = (m == k_y)       ? w_a : 0.0f;
    a2.x = (m == (8 + k_x)) ? w_b : 0.0f;
    a2.y = (m == (8 + k_y)) ? w_b : 0.0f;

    // --- Gather pointers (loop-invariant): B row k = h[src[edge k]].
    const float* __restrict__ row1x = hin + (size_t)src[base + k_x]     * D_FEAT;
    const float* __restrict__ row1y = hin + (size_t)src[base + k_y]     * D_FEAT;
    const float* __restrict__ row2x = hin + (size_t)src[base + 4 + k_x] * D_FEAT;
    const float* __restrict__ row2y = hin + (size_t)src[base + 4 + k_y] * D_FEAT;

    // --- Scatter pointers (loop-invariant): D VGPR r holds row r (lanes 0-15,
    // edge base+r) and row r+8 (lanes 16-31, edge base+4+r), column m.
    const int eoff = hi << 2;
    float* out0 = hout + (size_t)dst[base + 0 + eoff] * D_FEAT + m;
    float* out1 = hout + (size_t)dst[base + 1 + eoff] * D_FEAT + m;
    float* out2 = hout + (size_t)dst[base + 2 + eoff] * D_FEAT + m;
    float* out3 = hout + (size_t)dst[base + 3 + eoff] * D_FEAT + m;

#pragma unroll
    for (int t = 0; t < 4; ++t) {              // four 16-wide feature tiles
        const int f0 = t << 4;

        // B(4x16): row k striped across lanes, mirroring A's half-wave split.
        v2f b1, b2;
        b1.x = row1x[f0 + m];
        b1.y = row1y[f0 + m];
        b2.x = row2x[f0 + m];
        b2.y = row2y[f0 + m];

        v8f c = {};
        v8f d = __builtin_amdgcn_wmma_f32_16x16x4_f32(
            false, a1, false, b1, (short)0, c, false, false);
        d = __builtin_amdgcn_wmma_f32_16x16x4_f32(
            false, a2, false, b2, (short)0, d, false, false);

        // Full-wave fire-and-forget f32 atomics; f0 folds into the immediate
        // offset, base addresses hoisted out of the loop.
        atomicAdd(out0 + f0, d[0]);
        atomicAdd(out1 + f0, d[1]);
        atomicAdd(out2 + f0, d[2]);
        atomicAdd(out3 + f0, d[3]);
    }
}

extern "C" void kernel_launch(void* const* d_in, const int* in_sizes, int n_in,
                              void* d_out, int out_size, void* d_ws, size_t ws_size,
                              hipStream_t stream) {
    const float* x   = (const float*)d_in[0];
    const float* ew  = (const float*)d_in[1];
    const int*   src = (const int*)d_in[2];
    const int*   dst = (const int*)d_in[3];
    float* out = (float*)d_out;
    float* ws0 = (float*)d_ws;                 // 25.6 MB ping buffer

    const int n4          = N_NODES * D_FEAT / 4;     // 1.6M float4
    const int zero_blocks = (n4 + 255) / 256;
    const int hop_blocks  = N_EDGES / 64;             // 8 edges/wave, 8 waves/block

    // hop1: x -> out
    gcn_zero_kernel<<<zero_blocks, 256, 0, stream>>>((float4*)out, n4);
    gcn_hop_kernel<<<hop_blocks, 256, 0, stream>>>(x, out, src, dst, ew);
    // hop2: out -> ws0
    gcn_zero_kernel<<<zero_blocks, 256, 0, stream>>>((float4*)ws0, n4);
    gcn_hop_kernel<<<hop_blocks, 256, 0, stream>>>(out, ws0, src, dst, ew);
    // hop3: ws0 -> out
    gcn_zero_kernel<<<zero_blocks, 256, 0, stream>>>((float4*)out, n4);
    gcn_hop_kernel<<<hop_blocks, 256, 0, stream>>>(ws0, out, src, dst, ew);
}